// forward_warp_max_motion_83811991814576
// MI455X (gfx1250) — compile-verified
//
#include <hip/hip_runtime.h>
#include <cstdint>

namespace {

constexpr int B = 4, C = 3, H = 1080, W = 1920;
constexpr int HW   = H * W;        // 2,073,600 (multiple of 4)
constexpr int NPIX = B * HW;       // 8,294,400
constexpr int THREADS = 256;       // 8 wave32 waves per block

// ---- CDNA5 async global->LDS staging of the per-lane flow vector ----------
// Each lane DMAs its 8-byte float2 from global memory directly into LDS via
// GLOBAL_LOAD_ASYNC_TO_LDS_B64 (tracked with ASYNCcnt), letting us issue other
// global loads before S_WAIT_ASYNCCNT. Generic LDS pointers on gfx1250 carry
// the wave-relative LDS byte offset in their low 32 bits (ISA aperture rules),
// which is exactly what the instruction's VDST operand wants.
__device__ __forceinline__ void flow_async_issue(const float* __restrict__ flow,
                                                 int n, float2* sbuf) {
    uint32_t lds_off = (uint32_t)(uintptr_t)(&sbuf[threadIdx.x]);
    uint32_t goff    = (uint32_t)n * 8u;                 // byte offset, < 2^31
    unsigned long long base = (unsigned long long)(uintptr_t)flow;
    asm volatile("global_load_async_to_lds_b64 %0, %1, %2"
                 :: "v"(lds_off), "v"(goff), "s"(base)
                 : "memory");
}

__device__ __forceinline__ void flow_async_wait() {
    asm volatile("s_wait_asynccnt 0x0" ::: "memory");
}

// ---- 128-bit zero fill -----------------------------------------------------
__global__ __launch_bounds__(THREADS) void fw_zero(uint4* __restrict__ p, int n4) {
    int i = blockIdx.x * THREADS + threadIdx.x;
    if (i < n4) {
        uint4 z; z.x = 0u; z.y = 0u; z.z = 0u; z.w = 0u;
        p[i] = z;
    }
}

// ---- Pass 1: scatter-max depth (z-buffer) ---------------------------------
__global__ __launch_bounds__(THREADS) void fw_depth(const float* __restrict__ flow,
                                                    int* __restrict__ dbuf) {
    __shared__ float2 sflow[THREADS];
    const int n = blockIdx.x * THREADS + threadIdx.x;   // grid is exact: n < NPIX

    flow_async_issue(flow, n, sflow);
    flow_async_wait();
    const float2 f = sflow[threadIdx.x];

    const int b  = n / HW;
    const int hw = n - b * HW;
    const int y  = hw / W;
    const int x  = hw - y * W;

    const float xd = (float)x + f.x;
    const float yd = (float)y + f.y;
    const int   d  = (int)(100.0f * sqrtf(f.x * f.x + f.y * f.y));

    const float xf = floorf(xd), yf = floorf(yd);
    const int x_f = (int)xf, y_f = (int)yf;
    const int x_c = x_f + 1,  y_c = y_f + 1;
    if (x_f < 0 || x_c > W || y_f < 0 || y_c > H) return;   // 'valid' mask

    const float ax0 = xd - xf,            ay0 = yd - yf;            // frac
    const float ax1 = (xf + 1.0f) - xd,   ay1 = (yf + 1.0f) - yd;   // 1-frac

    const int xfc = min(max(x_f, 0), W - 1), yfc = min(max(y_f, 0), H - 1);
    const int xcc = min(max(x_c, 0), W - 1), ycc = min(max(y_c, 0), H - 1);
    const int base = b * HW;

    const float k0 = ax1 * ay1, k1 = ax0 * ay1, k2 = ax1 * ay0, k3 = ax0 * ay0;
    if (k0 >= 0.25f) atomicMax(&dbuf[base + yfc * W + xfc], d);
    if (k1 >= 0.25f) atomicMax(&dbuf[base + yfc * W + xcc], d);
    if (k2 >= 0.25f) atomicMax(&dbuf[base + ycc * W + xfc], d);
    if (k3 >= 0.25f) atomicMax(&dbuf[base + ycc * W + xcc], d);
}

// ---- Pass 2: depth test + bilinear scatter-add splat ----------------------
__global__ __launch_bounds__(THREADS) void fw_splat(const float* __restrict__ im0,
                                                    const float* __restrict__ flow,
                                                    const int*  __restrict__ dbuf,
                                                    float* __restrict__ wght,
                                                    float* __restrict__ out) {
    __shared__ float2 sflow[THREADS];
    const int n = blockIdx.x * THREADS + threadIdx.x;

    flow_async_issue(flow, n, sflow);            // async DMA of flow into LDS

    const int b  = n / HW;
    const int hw = n - b * HW;
    const int y  = hw / W;
    const int x  = hw - y * W;

    // Overlap: issue the three strided im0 channel loads while flow streams in.
    const size_t cbase = (size_t)b * (C * HW) + (size_t)hw;
    const float p0 = im0[cbase];
    const float p1 = im0[cbase + HW];
    const float p2 = im0[cbase + 2 * HW];

    flow_async_wait();
    const float2 f = sflow[threadIdx.x];

    const float xd = (float)x + f.x;
    const float yd = (float)y + f.y;
    const float df = (float)(int)(100.0f * sqrtf(f.x * f.x + f.y * f.y));

    const float xf = floorf(xd), yf = floorf(yd);
    const int x_f = (int)xf, y_f = (int)yf;
    const int x_c = x_f + 1,  y_c = y_f + 1;
    if (x_f < 0 || x_c > W || y_f < 0 || y_c > H) return;   // 'valid' mask

    const float ax0 = xd - xf,            ay0 = yd - yf;
    const float ax1 = (xf + 1.0f) - xd,   ay1 = (yf + 1.0f) - yd;

    const int xfc = min(max(x_f, 0), W - 1), yfc = min(max(y_f, 0), H - 1);
    const int xcc = min(max(x_c, 0), W - 1), ycc = min(max(y_c, 0), H - 1);
    const int base   = b * HW;
    const size_t ob  = (size_t)b * (C * HW);

    const float k[4]  = { ax1 * ay1, ax0 * ay1, ax1 * ay0, ax0 * ay0 };
    const int   sp[4] = { yfc * W + xfc, yfc * W + xcc,
                          ycc * W + xfc, ycc * W + xcc };

#pragma unroll
    for (int c4 = 0; c4 < 4; ++c4) {
        const int idx = base + sp[c4];
        const float dg = (float)dbuf[idx];                 // gather z-buffer
        if (k[c4] >= 0.25f && (dg - df) <= 25.0f) {        // MOTION_TH = 25.0
            const float w = k[c4];
            atomicAdd(&wght[idx], w);                      // non-returning adds
            atomicAdd(&out[ob + (size_t)sp[c4]],            w * p0);
            atomicAdd(&out[ob + HW + (size_t)sp[c4]],       w * p1);
            atomicAdd(&out[ob + 2 * HW + (size_t)sp[c4]],   w * p2);
        }
    }
}

// ---- Pass 3: normalize in place, 4 pixels / thread, all-b128 traffic ------
__global__ __launch_bounds__(THREADS) void fw_norm(const float* __restrict__ wght,
                                                   float* __restrict__ out) {
    const int q = blockIdx.x * THREADS + threadIdx.x;     // quad index, exact grid
    const int n = q * 4;                                  // first pixel of quad
    const int b  = n / HW;                                // quad never straddles b
    const int hw = n - b * HW;

    const float4 w4 = *(const float4*)(wght + n);
    float4 den;
    den.x = fmaxf(w4.x / 3.0f, 1e-5f);                    // wght_buf / C, clamp EPS
    den.y = fmaxf(w4.y / 3.0f, 1e-5f);
    den.z = fmaxf(w4.z / 3.0f, 1e-5f);
    den.w = fmaxf(w4.w / 3.0f, 1e-5f);

    float* o = out + (size_t)b * (C * HW) + (size_t)hw;
#pragma unroll
    for (int c = 0; c < C; ++c) {
        float4 v = *(float4*)(o + (size_t)c * HW);
        v.x = v.x / den.x;
        v.y = v.y / den.y;
        v.z = v.z / den.z;
        v.w = v.w / den.w;
        *(float4*)(o + (size_t)c * HW) = v;
    }
}

} // namespace

extern "C" void kernel_launch(void* const* d_in, const int* /*in_sizes*/, int /*n_in*/,
                              void* d_out, int out_size, void* d_ws, size_t /*ws_size*/,
                              hipStream_t stream) {
    const float* im0  = (const float*)d_in[0];   // [B,C,H,W] f32
    const float* flow = (const float*)d_in[1];   // [B,H,W,2] f32
    float* out = (float*)d_out;                  // [B,C,H,W] f32

    int*   dbuf = (int*)d_ws;                    // [NPIX] i32 z-buffer
    float* wght = (float*)(dbuf + NPIX);         // [NPIX] f32 weight accumulator

    // Zero z-buffer + weights (2*NPIX dwords) and the output accumulator.
    const int ws_n4  = (2 * NPIX) / 4;           // 4,147,200 -> exact
    const int out_n4 = out_size / 4;             // 6,220,800 -> exact
    fw_zero<<<(ws_n4  + THREADS - 1) / THREADS, THREADS, 0, stream>>>((uint4*)d_ws, ws_n4);
    fw_zero<<<(out_n4 + THREADS - 1) / THREADS, THREADS, 0, stream>>>((uint4*)out,  out_n4);

    const int nblk = NPIX / THREADS;             // 32,400 blocks, no tail
    fw_depth<<<nblk, THREADS, 0, stream>>>(flow, dbuf);
    fw_splat<<<nblk, THREADS, 0, stream>>>(im0, flow, dbuf, wght, out);
    fw_norm <<<NPIX / 4 / THREADS, THREADS, 0, stream>>>(wght, out);   // 8,100 blocks
}